// GraphSAGE_88888643158267
// MI455X (gfx1250) — compile-verified
//
#include <hip/hip_runtime.h>

typedef float v2f __attribute__((ext_vector_type(2)));
typedef float v8f __attribute__((ext_vector_type(8)));

#define D_H 128
#define LSTR 68   // 64 + 4 pad: bank = (4*row + k) % 64 -> conflict-free fragment reads

// ---------------- utility kernels ----------------

__global__ void zero_u32_kernel(unsigned int* p, long long n) {
    long long i = (long long)blockIdx.x * blockDim.x + threadIdx.x;
    if (i < n) p[i] = 0u;
}

__global__ void copy4_kernel(const float4* __restrict__ src, float4* __restrict__ dst, long long n4) {
    long long i = (long long)blockIdx.x * blockDim.x + threadIdx.x;
    if (i < n4) dst[i] = src[i];
}

// ---------------- graph aggregation ----------------

__global__ void degree_kernel(const long long* __restrict__ e0, const long long* __restrict__ e1,
                              unsigned int* __restrict__ deg, int E) {
    int i = blockIdx.x * blockDim.x + threadIdx.x;
    if (i < E) {
        atomicAdd(&deg[(int)e0[i]], 1u);
        atomicAdd(&deg[(int)e1[i]], 1u);
    }
}

__global__ void invdeg_kernel(const unsigned int* __restrict__ deg, float* __restrict__ invdeg, int n) {
    int i = blockIdx.x * blockDim.x + threadIdx.x;
    if (i < n) invdeg[i] = 1.0f / (float)(deg[i] + 1u);  // +1 self loop
}

// One edge-direction handled by 32 threads (4 floats each). A must be pre-initialized
// with the self-loop contribution (A = X).
__global__ void agg_scatter_kernel(const long long* __restrict__ e0, const long long* __restrict__ e1,
                                   const float* __restrict__ X, float* __restrict__ A, int E) {
    long long gid = (long long)blockIdx.x * blockDim.x + threadIdx.x;
    long long dir = gid >> 5;
    int q = (int)(gid & 31) << 2;
    if (dir >= 2LL * E) return;
    int u, v;
    if (dir < E) { u = (int)e0[dir]; v = (int)e1[dir]; }
    else         { u = (int)e1[dir - E]; v = (int)e0[dir - E]; }
    float4 val = *(const float4*)(X + (long long)u * D_H + q);
    float* dst = A + (long long)v * D_H + q;
    unsafeAtomicAdd(dst + 0, val.x);
    unsafeAtomicAdd(dst + 1, val.y);
    unsafeAtomicAdd(dst + 2, val.z);
    unsafeAtomicAdd(dst + 3, val.w);
}

// ---------------- WMMA GEMM: Out[r][n] = sum_k In[r][k] * W[n][k] (+bias) ----------------
// Block: 128 threads = 4 wave32. Block tile: 64 rows x 64 cols. K = 128.
// DUAL: accumulate second product (Agg * invdeg) @ Wneigh^T into the same tile.

template <bool DUAL>
__global__ __launch_bounds__(128)
void gemm_wmma_kernel(const float* __restrict__ X, const float* __restrict__ Wself,
                      const float* __restrict__ b1,
                      const float* __restrict__ Agg, const float* __restrict__ Wneigh,
                      const float* __restrict__ b2,
                      const float* __restrict__ invdeg,
                      float* __restrict__ Out, int Nrows, int out_ld) {
    __shared__ float xs[64 * LSTR];
    __shared__ float ws[64 * LSTR];

    const int tid  = threadIdx.x;
    const int lane = tid & 31;
    const int warp = tid >> 5;
    const int row0 = blockIdx.x * 64;
    const int nbase = blockIdx.y * 64;

    const int koff = (lane >> 4) << 1;              // 0 or 2 (half-wave K offset)
    const int mloc = warp * 16 + (lane & 15);       // this lane's M row in the 64-row tile
    const int nl   = lane & 15;

    v8f acc[4] = {};                                // 4 tiles of 16x16 over N

    const int nphase = DUAL ? 2 : 1;
    for (int phase = 0; phase < nphase; ++phase) {
        const float* src = (DUAL && phase) ? Agg : X;
        const float* W   = (DUAL && phase) ? Wneigh : Wself;
        for (int kc = 0; kc < 128; kc += 64) {
            __syncthreads();
            // cooperative load: 64x64 input tile + 64x64 weight tile (float4 coalesced)
            #pragma unroll
            for (int i = 0; i < 8; ++i) {
                int j  = tid + i * 128;
                int r  = j >> 4;
                int c4 = (j & 15) << 2;
                int grow = row0 + r;
                float4 v = make_float4(0.f, 0.f, 0.f, 0.f);
                if (grow < Nrows) {
                    v = *(const float4*)(src + (long long)grow * 128 + kc + c4);
                    if (DUAL && phase) {
                        float s = invdeg[grow];
                        v.x *= s; v.y *= s; v.z *= s; v.w *= s;
                    }
                }
                *(float4*)(xs + r * LSTR + c4) = v;
                float4 w = *(const float4*)(W + (long long)(nbase + r) * 128 + kc + c4);
                *(float4*)(ws + r * LSTR + c4) = w;
            }
            __syncthreads();
            // K loop: 16 WMMA steps of K=4 each
            for (int kk = 0; kk < 64; kk += 4) {
                v2f a = *(const v2f*)(xs + mloc * LSTR + kk + koff);
                #pragma unroll
                for (int nt = 0; nt < 4; ++nt) {
                    v2f b = *(const v2f*)(ws + (nt * 16 + nl) * LSTR + kk + koff);
                    acc[nt] = __builtin_amdgcn_wmma_f32_16x16x4_f32(
                        false, a, false, b, (short)0, acc[nt], false, false);
                }
            }
        }
    }

    // bias + store (C/D layout: reg r -> M = r + 8*(lane/16), N = lane%16)
    const int mhi = (lane >> 4) << 3;
    #pragma unroll
    for (int nt = 0; nt < 4; ++nt) {
        int ncol = nbase + nt * 16 + nl;
        float bias = b1[ncol] + (DUAL ? b2[ncol] : 0.0f);
        #pragma unroll
        for (int r = 0; r < 8; ++r) {
            int grow = row0 + warp * 16 + r + mhi;
            if (grow < Nrows) Out[(long long)grow * out_ld + ncol] = acc[nt][r] + bias;
        }
    }
}

// ---------------- BatchNorm (training-mode batch stats) ----------------

__global__ void stats_kernel(const float* __restrict__ C, float* __restrict__ stats, int N) {
    __shared__ float shs[256];
    __shared__ float shq[256];
    int col  = threadIdx.x & 127;
    int half = threadIdx.x >> 7;
    int row0 = blockIdx.x * 256;
    float sum = 0.f, sq = 0.f;
    for (int r = half; r < 256; r += 2) {
        int row = row0 + r;
        if (row < N) {
            float v = C[(long long)row * 128 + col];
            sum += v; sq += v * v;
        }
    }
    shs[threadIdx.x] = sum;
    shq[threadIdx.x] = sq;
    __syncthreads();
    if (half == 0) {
        unsafeAtomicAdd(&stats[col],       shs[col] + shs[col + 128]);
        unsafeAtomicAdd(&stats[128 + col], shq[col] + shq[col + 128]);
    }
}

__global__ void bnrelu_kernel(const float* __restrict__ C, const float* __restrict__ stats,
                              const float* __restrict__ gamma, const float* __restrict__ beta,
                              float* __restrict__ Out, long long total, float invN) {
    long long i = (long long)blockIdx.x * blockDim.x + threadIdx.x;
    if (i >= total) return;
    int col = (int)(i & 127);
    float mean = stats[col] * invN;
    float var  = stats[128 + col] * invN - mean * mean;
    float scl  = gamma[col] * rsqrtf(var + 1e-5f);
    float v = (C[i] - mean) * scl + beta[col];
    Out[i] = v > 0.f ? v : 0.f;
}

// ---------------- driver ----------------

extern "C" void kernel_launch(void* const* d_in, const int* in_sizes, int n_in,
                              void* d_out, int out_size, void* d_ws, size_t ws_size,
                              hipStream_t stream) {
    const float*     x       = (const float*)d_in[0];
    const long long* ei      = (const long long*)d_in[1];
    const float*     Wself1  = (const float*)d_in[2];
    const float*     bself1  = (const float*)d_in[3];
    const float*     Wneigh1 = (const float*)d_in[4];
    const float*     bneigh1 = (const float*)d_in[5];
    const float*     gamma1  = (const float*)d_in[6];
    const float*     beta1   = (const float*)d_in[7];
    const float*     Wself2  = (const float*)d_in[8];
    const float*     bself2  = (const float*)d_in[9];
    const float*     Wneigh2 = (const float*)d_in[10];
    const float*     bneigh2 = (const float*)d_in[11];
    const float*     gamma2  = (const float*)d_in[12];
    const float*     beta2   = (const float*)d_in[13];
    const float*     Whead   = (const float*)d_in[14];
    const float*     bhead   = (const float*)d_in[15];

    const int N = in_sizes[0] / 128;
    const int E = in_sizes[1] / 2;
    const long long* e0 = ei;
    const long long* e1 = ei + E;

    char* wsb = (char*)d_ws;
    unsigned int* deg    = (unsigned int*)wsb;                 // N u32
    float*        invdeg = (float*)(wsb + (size_t)N * 4);      // N f32
    float*        A      = (float*)(wsb + (size_t)N * 8);      // N*128 (aggregate / h2)
    float*        B      = A + (size_t)N * 128;                // N*128 (h1)
    float*        C      = B + (size_t)N * 128;                // N*128 (pre-BN)
    float*        stats  = C + (size_t)N * 128;                // 256 f32

    const long long tot   = (long long)N * 128;
    const long long tot4  = tot >> 2;
    const long long sthr  = 2LL * E * 32;

    dim3 g2((N + 63) / 64, 2);
    dim3 g1((N + 63) / 64, 1);

    // degree + invdeg
    zero_u32_kernel<<<(N + 255) / 256, 256, 0, stream>>>(deg, N);
    degree_kernel<<<(E + 255) / 256, 256, 0, stream>>>(e0, e1, deg, E);
    invdeg_kernel<<<(N + 255) / 256, 256, 0, stream>>>(deg, invdeg, N);

    // layer 1 aggregate: A = x (self loop) + scatter
    copy4_kernel<<<(tot4 + 255) / 256, 256, 0, stream>>>((const float4*)x, (float4*)A, tot4);
    agg_scatter_kernel<<<(sthr + 255) / 256, 256, 0, stream>>>(e0, e1, x, A, E);

    // layer 1 GEMM -> C, BN stats, BN+ReLU -> B
    gemm_wmma_kernel<true><<<g2, 128, 0, stream>>>(x, Wself1, bself1, A, Wneigh1, bneigh1,
                                                   invdeg, C, N, 128);
    zero_u32_kernel<<<1, 256, 0, stream>>>((unsigned int*)stats, 256);
    stats_kernel<<<(N + 255) / 256, 256, 0, stream>>>(C, stats, N);
    bnrelu_kernel<<<(tot + 255) / 256, 256, 0, stream>>>(C, stats, gamma1, beta1, B, tot, 1.0f / N);

    // layer 2 aggregate: A = B + scatter
    copy4_kernel<<<(tot4 + 255) / 256, 256, 0, stream>>>((const float4*)B, (float4*)A, tot4);
    agg_scatter_kernel<<<(sthr + 255) / 256, 256, 0, stream>>>(e0, e1, B, A, E);

    // layer 2 GEMM -> C, BN stats, BN+ReLU -> A (h2)
    gemm_wmma_kernel<true><<<g2, 128, 0, stream>>>(B, Wself2, bself2, A, Wneigh2, bneigh2,
                                                   invdeg, C, N, 128);
    zero_u32_kernel<<<1, 256, 0, stream>>>((unsigned int*)stats, 256);
    stats_kernel<<<(N + 255) / 256, 256, 0, stream>>>(C, stats, N);
    bnrelu_kernel<<<(tot + 255) / 256, 256, 0, stream>>>(C, stats, gamma2, beta2, A, tot, 1.0f / N);

    // head GEMM -> d_out [N x 64]
    gemm_wmma_kernel<false><<<g1, 128, 0, stream>>>(A, Whead, bhead, nullptr, nullptr, nullptr,
                                                    nullptr, (float*)d_out, N, 64);
}